// TaxonomyOneHot_39436389712070
// MI455X (gfx1250) — compile-verified
//
#include <hip/hip_runtime.h>
#include <hip/hip_bf16.h>
#include <cstddef>

// TaxonomyOneHot: out[n, :] = relu(W[ids0[n]] + W[50+ids1[n]] + W[550+ids2[n]] + W[5550+ids3[n]] + b)
// Memory-bound gather-sum. CDNA5 path: async global->LDS gathers (ASYNCcnt),
// double-buffered per wave; scalar id loads; NT streaming stores.

typedef float v2f __attribute__((ext_vector_type(2)));
typedef int   v4i __attribute__((vector_size(16)));   // matches builtin param type

#if defined(__gfx1250__) &&                                              \
    __has_builtin(__builtin_amdgcn_global_load_async_to_lds_b128) &&     \
    __has_builtin(__builtin_amdgcn_s_wait_asynccnt)
#define USE_ASYNC_LDS 1
typedef __attribute__((address_space(1))) v4i* gvec_ptr;   // global (AS1)
typedef __attribute__((address_space(3))) v4i* lvec_ptr;   // LDS (AS3)
#else
#define USE_ASYNC_LDS 0
#endif

#define WAVES_PER_BLOCK 8
#define THREADS_PER_BLOCK (WAVES_PER_BLOCK * 32)

__global__ __launch_bounds__(THREADS_PER_BLOCK) void
TaxonomyOneHot_gather_kernel(const int* __restrict__ ids0,
                             const int* __restrict__ ids1,
                             const int* __restrict__ ids2,
                             const int* __restrict__ ids3,
                             const float* __restrict__ W,
                             const float* __restrict__ b,
                             float* __restrict__ out,
                             int n) {
  const int lane  = threadIdx.x & 31;
  const int wave  = threadIdx.x >> 5;
  const int gwave = blockIdx.x * WAVES_PER_BLOCK + wave;
  const int wstride = gridDim.x * WAVES_PER_BLOCK;

  // Per-lane bias: 2 floats (64 floats / 32 lanes).
  const v2f bias = *(const v2f*)(b + 2 * lane);

#if USE_ASYNC_LDS
  // 8 waves x 2 buffers x (4 gathered rows of 64 floats) = 16 KB LDS.
  __shared__ float lds[WAVES_PER_BLOCK * 2 * 4 * 64];
  float* wbase = &lds[wave * (2 * 4 * 64)];

  // Issue the 4 gathered W-rows for output row r into buffer bsel.
  // Two b128 async instructions: lanes 0-15 carry one W row (16 lanes x 16B
  // = 256B), lanes 16-31 the other.
  auto issue = [&](int r, int bsel) {
    const int r0 = __builtin_amdgcn_readfirstlane(r);
    const int b0 = ids0[r0] * 64;            // offset 0
    const int b1 = (ids1[r0] + 50) * 64;     // offset 50
    const int b2 = (ids2[r0] + 550) * 64;    // offset 550
    const int b3 = (ids3[r0] + 5550) * 64;   // offset 5550
    const int sub = (lane >> 4) & 1;
    const int j   = (lane & 15) * 4;         // 4 floats = 16B per lane
    const float* gA = W + (sub ? b1 : b0) + j;
    const float* gB = W + (sub ? b3 : b2) + j;
    float* lA = wbase + bsel * 256 + sub * 64 + j;  // rows 0/1 of buffer
    float* lB = lA + 128;                            // rows 2/3 of buffer
    __builtin_amdgcn_global_load_async_to_lds_b128(
        (gvec_ptr)gA, (lvec_ptr)lA, 0, 0);
    __builtin_amdgcn_global_load_async_to_lds_b128(
        (gvec_ptr)gB, (lvec_ptr)lB, 0, 0);
  };

  int row = gwave;
  int buf = 0;
  if (row < n) issue(row, 0);
  while (row < n) {
    const int nxt = row + wstride;
    if (nxt < n) {
      issue(nxt, buf ^ 1);                     // prefetch next row's gathers
      __builtin_amdgcn_s_wait_asynccnt(2);     // wait for current buffer only
    } else {
      __builtin_amdgcn_s_wait_asynccnt(0);
    }
    const v2f* src = (const v2f*)(wbase + buf * 256);
    const v2f a0 = src[0 * 32 + lane];
    const v2f a1 = src[1 * 32 + lane];
    const v2f a2 = src[2 * 32 + lane];
    const v2f a3 = src[3 * 32 + lane];
    v2f h = a0 + a1 + a2 + a3 + bias;
    h.x = fmaxf(h.x, 0.0f);
    h.y = fmaxf(h.y, 0.0f);
    __builtin_nontemporal_store(h, (v2f*)(out + (size_t)row * 64) + lane);
    buf ^= 1;
    row = nxt;
  }
#else
  // Fallback: direct gathers to VGPRs (still bandwidth-optimal layout).
  for (int row = gwave; row < n; row += wstride) {
    const int r0 = __builtin_amdgcn_readfirstlane(row);
    const int b0 = ids0[r0] * 64;
    const int b1 = (ids1[r0] + 50) * 64;
    const int b2 = (ids2[r0] + 550) * 64;
    const int b3 = (ids3[r0] + 5550) * 64;
    const v2f a0 = *((const v2f*)(W + b0) + lane);
    const v2f a1 = *((const v2f*)(W + b1) + lane);
    const v2f a2 = *((const v2f*)(W + b2) + lane);
    const v2f a3 = *((const v2f*)(W + b3) + lane);
    v2f h = a0 + a1 + a2 + a3 + bias;
    h.x = fmaxf(h.x, 0.0f);
    h.y = fmaxf(h.y, 0.0f);
    __builtin_nontemporal_store(h, (v2f*)(out + (size_t)row * 64) + lane);
  }
#endif
}

extern "C" void kernel_launch(void* const* d_in, const int* in_sizes, int n_in,
                              void* d_out, int out_size, void* d_ws, size_t ws_size,
                              hipStream_t stream) {
  (void)n_in; (void)out_size; (void)d_ws; (void)ws_size;
  const int*   ids0 = (const int*)d_in[0];
  const int*   ids1 = (const int*)d_in[1];
  const int*   ids2 = (const int*)d_in[2];
  const int*   ids3 = (const int*)d_in[3];
  const float* W    = (const float*)d_in[4];
  const float* b    = (const float*)d_in[5];
  float*       out  = (float*)d_out;
  const int n = in_sizes[0];  // 200000 rows

  int blocks = (n + WAVES_PER_BLOCK - 1) / WAVES_PER_BLOCK;
  if (blocks > 2048) blocks = 2048;  // ~12 rows/wave -> pipeline amortization
  TaxonomyOneHot_gather_kernel<<<blocks, THREADS_PER_BLOCK, 0, stream>>>(
      ids0, ids1, ids2, ids3, W, b, out, n);
}